// AvgPoolingLayer_61589831024881
// MI455X (gfx1250) — compile-verified
//
#include <hip/hip_runtime.h>

// Segment-mean (fused segment_sum + counts + divide) for MI455X / gfx1250.
// feats: [N, 128] f32, segment_ids: [N] i32 sorted ascending, out: [S, 128] f32.
//
// Memory-bound: 1.02 GB read => ~44us floor at 23.3 TB/s. Each block owns 256
// contiguous rows; wave w owns columns [16w,16w+16) so every element is read
// exactly once. Segmented reduction on the matrix pipe:
//   C(16 local segs x 16 cols) += A(16x4 one-hot) x B(4 rows x 16 cols)
// via V_WMMA_F32_16X16X4_F32. All segment bookkeeping is precomputed once per
// block (prefix sum over boundary flags), so the inner loop is just
// ds_load + 2 global loads + indicator build + wmma.

typedef __attribute__((ext_vector_type(2))) float v2f;
typedef __attribute__((ext_vector_type(8))) float v8f;

#define SEG_D 128
#define CHUNK 256   // rows per block (== blockDim.x)

__global__ __launch_bounds__(256) void seg_zero(float* __restrict__ outsum,
                                                float* __restrict__ cnt,
                                                int total, int S) {
  int i = blockIdx.x * 256 + threadIdx.x;
  if (i < total) outsum[i] = 0.0f;
  if (i < S)     cnt[i]    = 0.0f;
}

struct SmemT {
  alignas(16) int seg[CHUNK];        // segment id per row (-1 past N)
  alignas(16) int lloc1[CHUNK];      // 1-based local segment index per row
  alignas(16) int first[CHUNK + 1];  // first row of each distinct segment + sentinel
};

// Hot loop, one wave = one 16-column tile. GUARD only for the single tail block.
template <bool GUARD>
__device__ __forceinline__ void accum_loop(const float* __restrict__ feats,
                                           float* __restrict__ outsum,
                                           const SmemT& sm, int ndist,
                                           long long row0, int N,
                                           int half, int nidx, int c0) {
  v8f acc;
#pragma unroll
  for (int v = 0; v < 8; ++v) acc[v] = 0.0f;

  int base1 = 1;                 // window start in 1-based lloc space
  const int ka = half << 1;      // A/B K pair handled by this lane half: {0,1} or {2,3}
  const int mrow = half << 3;    // C rows this lane holds: v+0 or v+8

  for (int rb = 0; rb < CHUNK; rb += 4) {
    const int4 l4 = *(const int4*)&sm.lloc1[rb];

    if (l4.w >= base1 + 16) {    // window full: flush & restart at this quad
#pragma unroll
      for (int v = 0; v < 8; ++v) {
        int d = (base1 - 1) + v + mrow;          // distinct-segment index
        if (d < ndist) {
          int g = sm.seg[sm.first[d]];           // global segment id
          atomicAdd(&outsum[(long long)g * SEG_D + c0], acc[v]);
        }
        acc[v] = 0.0f;
      }
      base1 = l4.x;              // lloc grows <=1/row, so quad fits after restart
    }

    // A: 16x4 one-hot indicator. lane = M, VGPR0 = K{0|2}, VGPR1 = K{1|3} by half.
    const int t = nidx + base1;                  // lm[k]==nidx  <=>  l4[k]==t
    const int la0 = ka ? l4.z : l4.x;
    const int la1 = ka ? l4.w : l4.y;
    v2f A;
    A[0] = (la0 == t) ? 1.0f : 0.0f;
    A[1] = (la1 == t) ? 1.0f : 0.0f;

    // B: 4x16 feature rows, lane = column; VGPR k-split assumed to mirror A.
    const long long rB0 = row0 + rb + ka;
    v2f B;
    if (GUARD) {
      B[0] = (rB0     < (long long)N) ? feats[ rB0      * SEG_D + c0] : 0.0f;
      B[1] = (rB0 + 1 < (long long)N) ? feats[(rB0 + 1) * SEG_D + c0] : 0.0f;
    } else {
      B[0] = feats[ rB0      * SEG_D + c0];
      B[1] = feats[(rB0 + 1) * SEG_D + c0];
    }

    acc = __builtin_amdgcn_wmma_f32_16x16x4_f32(
        /*neg_a=*/false, A, /*neg_b=*/false, B,
        /*c_mod=*/(short)0, acc, /*reuse_a=*/false, /*reuse_b=*/false);
  }

  // final flush
#pragma unroll
  for (int v = 0; v < 8; ++v) {
    int d = (base1 - 1) + v + mrow;
    if (d < ndist) {
      int g = sm.seg[sm.first[d]];
      atomicAdd(&outsum[(long long)g * SEG_D + c0], acc[v]);
    }
  }
}

__global__ __launch_bounds__(256) void seg_accum(const float* __restrict__ feats,
                                                 const int* __restrict__ segids,
                                                 float* __restrict__ outsum,
                                                 float* __restrict__ cnt,
                                                 int N) {
  __shared__ SmemT sm;
  __shared__ int s_scan[CHUNK];

  const int tid = threadIdx.x;
  const long long row0 = (long long)blockIdx.x * CHUNK;
  const long long rr = row0 + tid;
  const bool valid = rr < (long long)N;
  const int s = valid ? segids[rr] : -1;
  sm.seg[tid] = s;
  __syncthreads();

  // boundary flag -> block-wide inclusive prefix sum (Hillis-Steele, 8 steps)
  const int b = (valid && (tid == 0 || s != sm.seg[tid - 1])) ? 1 : 0;
  s_scan[tid] = b;
  __syncthreads();
#pragma unroll
  for (int ofs = 1; ofs < CHUNK; ofs <<= 1) {
    int v = s_scan[tid];
    int add = (tid >= ofs) ? s_scan[tid - ofs] : 0;
    __syncthreads();
    s_scan[tid] = v + add;
    __syncthreads();
  }

  const int ndist = s_scan[CHUNK - 1];           // distinct segments in chunk
  const long long rem = (long long)N - row0;
  const int nvalid = rem < CHUNK ? (int)rem : CHUNK;
  const int l1 = s_scan[tid];
  sm.lloc1[tid] = l1;
  if (b) sm.first[l1 - 1] = tid;
  if (tid == 0) sm.first[ndist] = nvalid;        // sentinel
  __syncthreads();

  // per-segment counts: one thread per distinct segment, one atomic each
  if (tid < ndist) {
    int f0 = sm.first[tid];
    int f1 = sm.first[tid + 1];
    atomicAdd(&cnt[sm.seg[f0]], (float)(f1 - f0));
  }

  const int lane = tid & 31;                     // wave32
  const int half = lane >> 4;
  const int nidx = lane & 15;
  const int wave = tid >> 5;
  const int c0 = wave * 16 + nidx;               // absolute feature column

  if (row0 + CHUNK <= (long long)N)
    accum_loop<false>(feats, outsum, sm, ndist, row0, N, half, nidx, c0);
  else
    accum_loop<true >(feats, outsum, sm, ndist, row0, N, half, nidx, c0);
}

__global__ __launch_bounds__(256) void seg_div(float* __restrict__ outsum,
                                               const float* __restrict__ cnt,
                                               int total) {
  int i = blockIdx.x * 256 + threadIdx.x;
  if (i < total) {
    int sidx = i >> 7;  // / 128
    outsum[i] = outsum[i] / fmaxf(cnt[sidx], 1.0f);
  }
}

extern "C" void kernel_launch(void* const* d_in, const int* in_sizes, int n_in,
                              void* d_out, int out_size, void* d_ws, size_t ws_size,
                              hipStream_t stream) {
  (void)n_in; (void)ws_size;
  const float* feats  = (const float*)d_in[0];
  const int*   segids = (const int*)d_in[1];
  const int N = in_sizes[0] / SEG_D;      // 2,000,000
  const int S = out_size / SEG_D;         // 16,384

  float* outsum = (float*)d_out;          // accumulate sums in place, divide after
  float* cnt    = (float*)d_ws;           // S floats of scratch (64 KB)

  const int zgrid = (out_size + 255) / 256;
  seg_zero<<<zgrid, 256, 0, stream>>>(outsum, cnt, out_size, S);

  const int agrid = (N + CHUNK - 1) / CHUNK;
  seg_accum<<<agrid, 256, 0, stream>>>(feats, segids, outsum, cnt, N);

  seg_div<<<zgrid, 256, 0, stream>>>(outsum, cnt, out_size);
}